// VQBottleneckLSTM_53566832115847
// MI455X (gfx1250) — compile-verified
//
#include <hip/hip_runtime.h>
#include <hip/hip_bf16.h>

typedef float v2f __attribute__((ext_vector_type(2)));
typedef float v8f __attribute__((ext_vector_type(8)));

#define BB 512
#define TT 2048
#define NBT (BB*TT)

// branch-free activations (v_exp_f32 / v_rcp_f32; saturate correctly via inf/0)
__device__ __forceinline__ float sigm_f(float x) {
    return __builtin_amdgcn_rcpf(1.0f + __expf(-x));
}
__device__ __forceinline__ float tanh_f(float x) {
    return 1.0f - 2.0f * __builtin_amdgcn_rcpf(__expf(2.0f * x) + 1.0f);
}

// ---------------- init: zero accumulators (ws is poisoned 0xAA) -------------
__global__ __launch_bounds__(32) void vq_init_kernel(float* wf) {
    if (threadIdx.x < 17) wf[threadIdx.x] = 0.0f;   // [0]=sq_err, [1..16]=counts
}

// ------- fused conv1d + BN + ReLU + proj + VQ argmin + loss/hist ------------
__global__ __launch_bounds__(256) void vq_frontend_kernel(
    const float* __restrict__ x,
    const float* __restrict__ conv_w, const float* __restrict__ conv_b,
    const float* __restrict__ bn_gamma, const float* __restrict__ bn_beta,
    const float* __restrict__ bn_mean, const float* __restrict__ bn_var,
    const float* __restrict__ proj_w, const float* __restrict__ proj_b,
    const float* __restrict__ codebook,
    unsigned char* __restrict__ idx_out,
    float* __restrict__ sq_err_acc, float* __restrict__ counts)
{
    __shared__ float s_cw[60];
    __shared__ float s_cb[4];
    __shared__ float s_bnA[4], s_bnB[4];
    __shared__ float s_pw[32], s_pb[8];
    __shared__ float s_code[128], s_cn[16];
    __shared__ unsigned int s_hist[16];
    __shared__ float s_err;

    int tid = threadIdx.x;
    if (tid < 60) s_cw[tid] = conv_w[tid];
    if (tid < 4) {
        s_cb[tid] = conv_b[tid];
        float a = bn_gamma[tid] * rsqrtf(bn_var[tid] + 1e-5f);
        s_bnA[tid] = a;
        s_bnB[tid] = bn_beta[tid] - bn_mean[tid] * a;
    }
    if (tid >= 64 && tid < 96)   s_pw[tid-64]    = proj_w[tid-64];
    if (tid >= 96 && tid < 104)  s_pb[tid-96]    = proj_b[tid-96];
    if (tid >= 128)              s_code[tid-128] = codebook[tid-128];
    if (tid < 16) s_hist[tid] = 0u;
    if (tid == 63) s_err = 0.0f;
    __syncthreads();
    if (tid < 16) {
        float s = 0.f;
        #pragma unroll
        for (int e = 0; e < 8; ++e) { float v = s_code[tid*8+e]; s += v*v; }
        s_cn[tid] = s;
    }
    __syncthreads();

    unsigned int u = blockIdx.x * 256u + tid;          // exactly B*T threads
    int t = (int)(u & (TT-1));
    const float* xp = x + (size_t)u * 5u;

    float xm[3][5];
    #pragma unroll
    for (int i = 0; i < 5; ++i) xm[1][i] = xp[i];
    #pragma unroll
    for (int i = 0; i < 5; ++i) xm[0][i] = (t > 0)    ? xp[i-5] : 0.f;
    #pragma unroll
    for (int i = 0; i < 5; ++i) xm[2][i] = (t < TT-1) ? xp[i+5] : 0.f;

    float flat[8];
    {
        float hb[4];
        #pragma unroll
        for (int c = 0; c < 4; ++c) {
            float acc = s_cb[c];
            #pragma unroll
            for (int i = 0; i < 5; ++i)
                #pragma unroll
                for (int dk = 0; dk < 3; ++dk)
                    acc += xm[dk][i] * s_cw[c*15 + i*3 + dk];
            acc = acc * s_bnA[c] + s_bnB[c];
            hb[c] = fmaxf(acc, 0.0f);
        }
        #pragma unroll
        for (int e = 0; e < 8; ++e) {
            float acc = s_pb[e];
            #pragma unroll
            for (int c = 0; c < 4; ++c) acc += hb[c] * s_pw[e*4+c];
            flat[e] = acc;
        }
    }

    // argmin_k ||c_k||^2 - 2 f.c_k  (||f||^2 constant over k; first-min tiebreak)
    int best = 0; float bestd = 3.4e38f;
    #pragma unroll
    for (int k = 0; k < 16; ++k) {
        float dot = 0.f;
        #pragma unroll
        for (int e = 0; e < 8; ++e) dot += flat[e] * s_code[k*8+e];
        float d = s_cn[k] - 2.0f * dot;
        if (d < bestd) { bestd = d; best = k; }
    }
    float err = 0.f;
    #pragma unroll
    for (int e = 0; e < 8; ++e) { float dv = s_code[best*8+e] - flat[e]; err += dv*dv; }

    idx_out[u] = (unsigned char)best;
    atomicAdd(&s_hist[best], 1u);
    atomicAdd(&s_err, err);
    __syncthreads();
    if (tid == 0)  atomicAdd(sq_err_acc, s_err);
    if (tid < 16)  atomicAdd(&counts[tid], (float)s_hist[tid]);
}

// ---- permuted gate table -----------------------------------------------------
// gt_perm[k][j][hi][r] = codebook[k].w_ih[g] + b_ih[g] + b_hh[g],
//   g = 16*j + (r>>1)*4 + 2*hi + (r&1)
// This row order makes the LSTM D-tile lane layout == next-step B layout.
__global__ __launch_bounds__(256) void vq_gate_table_kernel(
    const float* __restrict__ codebook, const float* __restrict__ w_ih,
    const float* __restrict__ b_ih, const float* __restrict__ b_hh,
    float* __restrict__ gt)
{
    for (int e0 = threadIdx.x; e0 < 1024; e0 += 256) {
        int k  = e0 >> 6;
        int j  = (e0 >> 4) & 3;
        int hi = (e0 >> 3) & 1;
        int r  = e0 & 7;
        int g  = 16*j + (r >> 1)*4 + 2*hi + (r & 1);
        float acc = b_ih[g] + b_hh[g];
        #pragma unroll
        for (int e = 0; e < 8; ++e) acc += codebook[k*8+e] * w_ih[g*8+e];
        gt[e0] = acc;
    }
}

// ------------- LSTM recurrence: WMMA f32 16x16x4, 16 batches / wave ----------
// G^T = W_hh_perm (A) x h^T (B) + gate_table_perm[idx] (C).
// Row permutation q(M) = ((M&7)>>1)*4 + 2*(M>>3) + (M&1) makes each lane's
// D registers hold exactly the hid units its next-step B operand needs:
//   Bc[kk] = { h[2kk], h[2kk+1] }   (no cross-lane traffic in the recurrence)
__global__ __launch_bounds__(32) void vq_lstm_kernel(
    const float* __restrict__ w_hh, const float* __restrict__ gate_table,
    const unsigned char* __restrict__ idx, float* __restrict__ out)
{
    __shared__ float gt[1024];
    int lane = threadIdx.x;
    for (int u2 = lane; u2 < 1024; u2 += 32) gt[u2] = gate_table[u2];
    __syncthreads();

    int m  = lane & 15;
    int hi = lane >> 4;            // lane half
    int b0 = blockIdx.x * 16;

    // A = permuted W_hh rows, WMMA A-layout: lane holds row m, comps K=4kk+2hi
    int qrow = ((m & 7) >> 1)*4 + 2*(m >> 3) + (m & 1);   // hid unit of A row m
    v2f A[4][4];
    #pragma unroll
    for (int j = 0; j < 4; ++j)
        #pragma unroll
        for (int kk = 0; kk < 4; ++kk) {
            int g  = 16*j + qrow;
            int k0 = 4*kk + 2*hi;
            A[j][kk].x = w_hh[g*16 + k0];
            A[j][kk].y = w_hh[g*16 + k0 + 1];
        }

    // state: lane l, reg r  <->  (batch m, hid q(r,hi) = (r>>1)*4 + 2hi + (r&1))
    float h[8], c[8];
    #pragma unroll
    for (int r = 0; r < 8; ++r) { h[r] = 0.f; c[r] = 0.f; }

    const unsigned int* ib32 =
        (const unsigned int*)(idx + (size_t)(b0 + m) * TT);
    int cbase = hi * 8;            // lane-half offset into permuted table row

    for (int t4 = 0; t4 < TT/4; ++t4) {
        unsigned int pack = ib32[t4];          // 4 codebook ids, one load
        #pragma unroll
        for (int s = 0; s < 4; ++s) {
            int base = (int)((pack >> (8*s)) & 255u) * 64 + cbase;

            v2f Bc[4];
            #pragma unroll
            for (int kk = 0; kk < 4; ++kk) { Bc[kk].x = h[2*kk]; Bc[kk].y = h[2*kk+1]; }

            v8f D[4];
            #pragma unroll
            for (int j = 0; j < 4; ++j) {          // tiles: i, f, g, o gates
                v8f acc;
                #pragma unroll
                for (int r = 0; r < 8; ++r) acc[r] = gt[base + 16*j + r];
                #pragma unroll
                for (int kk = 0; kk < 4; ++kk)
                    acc = __builtin_amdgcn_wmma_f32_16x16x4_f32(
                            false, A[j][kk], false, Bc[kk], (short)0, acc, false, false);
                D[j] = acc;
            }

            #pragma unroll
            for (int r = 0; r < 8; ++r) {
                float ig = sigm_f(D[0][r]);
                float fg = sigm_f(D[1][r]);
                float gg = tanh_f(D[2][r]);
                float og = sigm_f(D[3][r]);
                float cn = fg * c[r] + ig * gg;
                c[r] = cn;
                h[r] = og * tanh_f(cn);
            }
        }
    }

    #pragma unroll
    for (int r = 0; r < 8; ++r) {
        int q  = (r >> 1)*4 + 2*hi + (r & 1);
        int ob = (b0 + m)*16 + q;
        out[ob]        = h[r];     // output 0: h_last
        out[8192 + ob] = h[r];     // output 1: h_last (duplicate)
    }
}

// ------------------- finalize: vq_loss + perplexity -------------------------
__global__ __launch_bounds__(32) void vq_finalize_kernel(
    const float* __restrict__ wf, float* __restrict__ out)
{
    if (threadIdx.x == 0) {
        const float n = (float)NBT;
        float mse = wf[0] / (n * 8.0f);                 // e_loss == q_loss
        out[16384] = fminf(1.01f * mse, 5.0f);          // q + 0.01*e, clamp 5
        float ent = 0.f;
        for (int k = 0; k < 16; ++k) {
            float p = wf[1+k] / n;
            ent += p * logf(p + 1e-10f);
        }
        out[16385] = __expf(-ent);
    }
}

extern "C" void kernel_launch(void* const* d_in, const int* in_sizes, int n_in,
                              void* d_out, int out_size, void* d_ws, size_t ws_size,
                              hipStream_t stream) {
    const float* x        = (const float*)d_in[0];
    const float* conv_w   = (const float*)d_in[1];
    const float* conv_b   = (const float*)d_in[2];
    const float* bn_gamma = (const float*)d_in[3];
    const float* bn_beta  = (const float*)d_in[4];
    const float* bn_mean  = (const float*)d_in[5];
    const float* bn_var   = (const float*)d_in[6];
    const float* proj_w   = (const float*)d_in[7];
    const float* proj_b   = (const float*)d_in[8];
    const float* codebook = (const float*)d_in[9];
    const float* w_ih     = (const float*)d_in[10];
    const float* w_hh     = (const float*)d_in[11];
    const float* b_ih     = (const float*)d_in[12];
    const float* b_hh     = (const float*)d_in[13];
    float* out = (float*)d_out;

    float* wf     = (float*)d_ws;
    float* sq_err = wf;            // wf[0]
    float* counts = wf + 1;        // wf[1..16]
    float* gt     = wf + 32;       // 1024 floats (4 KB), permuted layout
    unsigned char* idxb = (unsigned char*)d_ws + 8192;   // B*T bytes

    vq_init_kernel<<<1, 32, 0, stream>>>(wf);
    vq_frontend_kernel<<<NBT/256, 256, 0, stream>>>(
        x, conv_w, conv_b, bn_gamma, bn_beta, bn_mean, bn_var,
        proj_w, proj_b, codebook, idxb, sq_err, counts);
    vq_gate_table_kernel<<<1, 256, 0, stream>>>(codebook, w_ih, b_ih, b_hh, gt);
    vq_lstm_kernel<<<BB/16, 32, 0, stream>>>(w_hh, gt, idxb, out);
    vq_finalize_kernel<<<1, 32, 0, stream>>>(wf, out);
}